// Involution_66477503807613
// MI455X (gfx1250) — compile-verified
//
#include <hip/hip_runtime.h>

typedef __attribute__((ext_vector_type(2))) float v2f;
typedef __attribute__((ext_vector_type(8))) float v8f;

// Problem constants (match reference setup_inputs)
constexpr int Bn   = 4;
constexpr int C    = 256;
constexpr int CRED = 64;    // C / reduction_ratio
constexpr int G    = 4;
constexpr int D    = 64;    // C / G
constexpr int K    = 7;
constexpr int PAD  = 3;
constexpr int H    = 56;
constexpr int W    = 56;
constexpr int HW   = H * W;       // 3136
constexpr int KK   = 49;          // K*K
constexpr int KKG  = 196;         // K*K*G
constexpr int NT   = HW / 16;     // 196 pixel tiles of 16 per batch
constexpr int NP   = NT / 2;      // 98 pairs of N tiles (one pair per wave)
constexpr float BN_EPS = 1e-5f;

// ---------------------------------------------------------------------------
// Kernel 1: r = ReLU(BN(w_reduce @ x + b_reduce))     [64 x 3136] per batch
// One wave computes a 16(M) x 32(N) tile: the A fragment (b64 load) is reused
// across two 16x16 WMMAs per K-step, halving A traffic and VMEM-per-wmma.
// A (16x4) layout: lanes 0-15 -> K=0,1 in v[0],v[1]; lanes 16-31 -> K=2,3.
// B (4x16) mirrored; C/D: VGPR v <-> row v+8*half, col = lane&15.
// ---------------------------------------------------------------------------
__global__ __launch_bounds__(256)
void involution_reduce_gemm(const float* __restrict__ x,
                            const float* __restrict__ wr,    // [64][256]
                            const float* __restrict__ br,
                            const float* __restrict__ gamma,
                            const float* __restrict__ beta,
                            const float* __restrict__ mean,
                            const float* __restrict__ var,
                            float* __restrict__ r)           // [B][64][HW]
{
  const int wave = (blockIdx.x << 3) + (threadIdx.x >> 5);
  const int lane = threadIdx.x & 31;
  const int half = lane >> 4;
  const int l16  = lane & 15;

  // tiles: b(4) x mt(4) x np(98) = 1568 waves total
  const int b  = wave / (4 * NP);
  const int t  = wave % (4 * NP);
  const int mt = t / NP;
  const int np = t % NP;

  const int m   = mt * 16 + l16;          // A row (output channel)
  const int hw0 = np * 32 + l16;          // B col, tile 0
  const float* Xb = x + (size_t)b * C * HW;

  v8f acc0 = {};
  v8f acc1 = {};
#pragma unroll 4
  for (int k0 = 0; k0 < C; k0 += 4) {
    const int ka = k0 + 2 * half;
    v2f a = *(const v2f*)(wr + m * C + ka);        // b64 load, consecutive K
    const float* p0 = Xb + (size_t)ka * HW + hw0;
    v2f b0, b1;
    b0.x = p0[0];       b0.y = p0[HW];             // tile 0 (cols hw0)
    b1.x = p0[16];      b1.y = p0[HW + 16];        // tile 1 (cols hw0+16)
    acc0 = __builtin_amdgcn_wmma_f32_16x16x4_f32(false, a, false, b0,
                                                 (short)0, acc0, false, false);
    acc1 = __builtin_amdgcn_wmma_f32_16x16x4_f32(false, a, false, b1,
                                                 (short)0, acc1, false, false);
  }

  float* Rb = r + (size_t)b * CRED * HW;
#pragma unroll
  for (int v = 0; v < 8; ++v) {
    const int o    = mt * 16 + v + 8 * half;
    const float sc = gamma[o] * rsqrtf(var[o] + BN_EPS);
    const float sh = beta[o] - mean[o] * sc;
    float v0 = fmaf(acc0[v] + br[o], sc, sh);
    float v1 = fmaf(acc1[v] + br[o], sc, sh);
    Rb[(size_t)o * HW + np * 32 + l16]      = fmaxf(v0, 0.0f);   // ReLU
    Rb[(size_t)o * HW + np * 32 + 16 + l16] = fmaxf(v1, 0.0f);
  }
}

// ---------------------------------------------------------------------------
// Kernel 2: ker = w_span @ r + b_span                [196 x 3136] per batch
// M = 196 -> 13 tiles of 16 (rows >=196 zero-padded / masked on store).
// Same 16x32-per-wave A-reuse scheme.
// ---------------------------------------------------------------------------
__global__ __launch_bounds__(256)
void involution_span_gemm(const float* __restrict__ r,       // [B][64][HW]
                          const float* __restrict__ ws,      // [196][64]
                          const float* __restrict__ bs,
                          float* __restrict__ ker)           // [B][196][HW]
{
  const int wave = (blockIdx.x << 3) + (threadIdx.x >> 5);
  const int lane = threadIdx.x & 31;
  const int half = lane >> 4;
  const int l16  = lane & 15;

  constexpr int MT2 = 13;                 // ceil(196/16)
  const int b  = wave / (MT2 * NP);
  const int t  = wave % (MT2 * NP);
  const int mt = t / NP;
  const int np = t % NP;

  const int m   = mt * 16 + l16;
  const int hw0 = np * 32 + l16;
  const float* Rb = r + (size_t)b * CRED * HW;

  v8f acc0 = {};
  v8f acc1 = {};
#pragma unroll
  for (int k0 = 0; k0 < CRED; k0 += 4) {
    const int ka = k0 + 2 * half;
    v2f a;
    if (m < KKG) {
      a = *(const v2f*)(ws + m * CRED + ka);
    } else {
      a.x = 0.0f; a.y = 0.0f;
    }
    const float* p0 = Rb + (size_t)ka * HW + hw0;
    v2f b0, b1;
    b0.x = p0[0];       b0.y = p0[HW];
    b1.x = p0[16];      b1.y = p0[HW + 16];
    acc0 = __builtin_amdgcn_wmma_f32_16x16x4_f32(false, a, false, b0,
                                                 (short)0, acc0, false, false);
    acc1 = __builtin_amdgcn_wmma_f32_16x16x4_f32(false, a, false, b1,
                                                 (short)0, acc1, false, false);
  }

  float* Kb = ker + (size_t)b * KKG * HW;
#pragma unroll
  for (int v = 0; v < 8; ++v) {
    const int o2 = mt * 16 + v + 8 * half;
    if (o2 < KKG) {
      const float bias = bs[o2];
      Kb[(size_t)o2 * HW + np * 32 + l16]      = acc0[v] + bias;
      Kb[(size_t)o2 * HW + np * 32 + 16 + l16] = acc1[v] + bias;
    }
  }
}

// ---------------------------------------------------------------------------
// Kernel 3: per-pixel dynamic depthwise 7x7 conv.
// Block = (8x8 pixel tile) x (one group) x (one batch); 256 threads =
// 64 pixels x 4 channel-quarters (16 channels each). LDS stages the
// 14x14x64-channel halo tile of x and the 64x49 per-pixel filters.
// out[b, g*64+c, h, w] = sum_kj x[b, g*64+c, h+kh-3, w+kw-3] * ker[b, kj*4+g, h, w]
// ---------------------------------------------------------------------------
__global__ __launch_bounds__(256)
void involution_apply(const float* __restrict__ x,
                      const float* __restrict__ ker,
                      float* __restrict__ out)
{
  extern __shared__ float smem[];
  float* sx = smem;              // [64][14*14] = 12544 floats
  float* sk = smem + D * 196;    // [64][49]    =  3136 floats

  const int tile = blockIdx.x;           // 0..48
  const int g    = blockIdx.y;           // 0..3
  const int b    = blockIdx.z;           // 0..3
  const int ty0  = (tile / 7) * 8;
  const int tx0  = (tile % 7) * 8;
  const int tid  = threadIdx.x;

  // Stage x halo tile for this group's 64 channels (zero-padded borders).
  const float* Xg = x + ((size_t)b * C + (size_t)g * D) * HW;
  for (int i = tid; i < D * 196; i += 256) {
    const int c   = i / 196;
    const int rem = i % 196;
    const int ry  = rem / 14;
    const int rx  = rem % 14;
    const int gy  = ty0 - PAD + ry;
    const int gx  = tx0 - PAD + rx;
    float v = 0.0f;
    if ((unsigned)gy < (unsigned)H && (unsigned)gx < (unsigned)W)
      v = Xg[(size_t)c * HW + gy * W + gx];
    sx[i] = v;
  }

  // Stage the 49 filter taps for each of the 64 pixels (o2 = kj*G + g).
  const float* Kb = ker + (size_t)b * KKG * HW;
  for (int i = tid; i < 64 * KK; i += 256) {
    const int p  = i / KK;
    const int kj = i % KK;
    const int gy = ty0 + (p >> 3);
    const int gx = tx0 + (p & 7);
    sk[i] = Kb[(size_t)(kj * G + g) * HW + gy * W + gx];
  }
  __syncthreads();

  const int p  = tid & 63;
  const int cs = (tid >> 6) * 16;        // channel-quarter base within group
  const int py = p >> 3;
  const int px = p & 7;

  float acc[16];
#pragma unroll
  for (int cc = 0; cc < 16; ++cc) acc[cc] = 0.0f;

  const float* skp = sk + p * KK;
#pragma unroll 7
  for (int kj = 0; kj < KK; ++kj) {
    const float kv = skp[kj];
    const int kh = kj / K;
    const int kw = kj % K;
    const float* sxp = sx + (py + kh) * 14 + (px + kw);
#pragma unroll
    for (int cc = 0; cc < 16; ++cc)
      acc[cc] = fmaf(sxp[(size_t)(cs + cc) * 196], kv, acc[cc]);
  }

  float* Ob = out + ((size_t)b * C + (size_t)(g * D + cs)) * HW
                  + (ty0 + py) * W + (tx0 + px);
#pragma unroll
  for (int cc = 0; cc < 16; ++cc)
    Ob[(size_t)cc * HW] = acc[cc];
}

// ---------------------------------------------------------------------------
extern "C" void kernel_launch(void* const* d_in, const int* in_sizes, int n_in,
                              void* d_out, int out_size, void* d_ws, size_t ws_size,
                              hipStream_t stream) {
  (void)in_sizes; (void)n_in; (void)out_size; (void)ws_size;

  const float* x  = (const float*)d_in[0];
  const float* wr = (const float*)d_in[1];
  const float* br = (const float*)d_in[2];
  const float* gm = (const float*)d_in[3];
  const float* bt = (const float*)d_in[4];
  const float* mn = (const float*)d_in[5];
  const float* vr = (const float*)d_in[6];
  const float* ws = (const float*)d_in[7];
  const float* bs = (const float*)d_in[8];
  float* out = (float*)d_out;

  // Workspace: r [4][64][3136] (3.06 MB) then ker [4][196][3136] (9.38 MB).
  float* r   = (float*)d_ws;
  float* ker = r + (size_t)Bn * CRED * HW;

  // GEMM1: 4*4*98 = 1568 waves, 8 waves/block -> 196 blocks.
  involution_reduce_gemm<<<196, 256, 0, stream>>>(x, wr, br, gm, bt, mn, vr, r);
  // GEMM2: 4*13*98 = 5096 waves -> 637 blocks.
  involution_span_gemm<<<637, 256, 0, stream>>>(r, ws, bs, ker);
  // Involution: 49 tiles x 4 groups x 4 batches; 62720 B dynamic LDS.
  involution_apply<<<dim3(49, 4, 4), 256, (D * 196 + D * KK) * sizeof(float),
                     stream>>>(x, ker, out);
}